// HermiteConvolution_46926812677022
// MI455X (gfx1250) — compile-verified
//
#include <hip/hip_runtime.h>
#include <cstddef>
#include <math.h>

#define IN_CHN 256
#define ORD    8
#define KW     9
#define BATCH  32
#define LEN    8192
#define NPOS   (BATCH*LEN)        // 262144
#define PI_F   3.14159265358979323846f

typedef __attribute__((ext_vector_type(2))) float v2f;
typedef __attribute__((ext_vector_type(8))) float v8f;

// ---- workspace layout (float offsets), total ~67KB ----
#define WS_P1     0        // 256 blocks * 32 stats
#define WS_P2     8192     // 256 blocks * 16 stats
#define WS_BN     12288    // 48 BN params (+pad)
#define WS_P3     12352    // 32b * 8lb * 8c
#define WS_WIDTH  14400    // 32
#define WS_CENTER 14432    // 32
#define WS_HERM   14464    // 32 * 72

// =====================================================================
// Kernel 1: x1 = W(8x256) . x(256 x L) per batch, via fp32 WMMA 16x16x4.
// Each wave owns a 32-column L tile (two 16x16 accumulators sharing one
// A fragment per K step: 1 ds_load_b64 -> 2 WMMAs).
// =====================================================================
__global__ void k_conv1(const float* __restrict__ x,
                        const float* __restrict__ w,
                        const float* __restrict__ bias,
                        float* __restrict__ x1)
{
    __shared__ float Wl[16 * IN_CHN];   // 16KB: rows 0..7 = W, rows 8..15 = 0
    const int tid = threadIdx.x;
    for (int i = tid; i < 16 * IN_CHN; i += blockDim.x)
        Wl[i] = (i < ORD * IN_CHN) ? w[i] : 0.0f;
    __syncthreads();

    const int wid  = tid >> 5;
    const int lane = tid & 31;
    const int tile = blockIdx.x * (blockDim.x >> 5) + wid; // 8192 tiles total
    const int b    = tile >> 8;            // 256 tiles per batch
    const int l0   = (tile & 255) << 5;    // 32 columns per tile
    const float* xb = x + (size_t)b * IN_CHN * LEN + l0;

    const int n  = lane & 15;              // N col (B/C/D) and M row (A)
    const int ko = (lane >> 4) << 1;       // K sub-offset: 0 or 2

    v8f acc0 = {0.f, 0.f, 0.f, 0.f, 0.f, 0.f, 0.f, 0.f};
    v8f acc1 = {0.f, 0.f, 0.f, 0.f, 0.f, 0.f, 0.f, 0.f};

#pragma unroll 8
    for (int k = 0; k < IN_CHN; k += 4) {
        const int ka = k + ko;
        // A fragment (16x4 of W) from LDS: lane holds A[n][ka], A[n][ka+1]
        v2f a = *(const v2f*)(&Wl[n * IN_CHN + ka]);
        // B fragments (4x16 of x): VGPR0 = row ka, VGPR1 = row ka+1.
        // Tile 1 is the same address + 64B -> folds into immediate offsets.
        const float* base = xb + (size_t)ka * LEN + n;
        v2f b0, b1;
        b0.x = base[0];
        b0.y = base[LEN];
        b1.x = base[16];
        b1.y = base[LEN + 16];
        acc0 = __builtin_amdgcn_wmma_f32_16x16x4_f32(false, a, false, b0,
                                                     (short)0, acc0, false, false);
        acc1 = __builtin_amdgcn_wmma_f32_16x16x4_f32(false, a, false, b1,
                                                     (short)0, acc1, false, false);
    }

    // D layout: VGPR r -> M=r (lanes 0-15), M=r+8 (lanes 16-31). Rows 8..15 unused.
    if (lane < 16) {
        float* o = x1 + (size_t)b * ORD * LEN + l0 + n;
#pragma unroll
        for (int r = 0; r < ORD; ++r) {
            const float br = bias[r];
            o[(size_t)r * LEN]      = acc0[r] + br;
            o[(size_t)r * LEN + 16] = acc1[r] + br;
        }
    }
}

// =====================================================================
// Helpers for the tiny 8-channel adaptive_r networks
// =====================================================================
__device__ __forceinline__ float waveSum(float v) {
#pragma unroll
    for (int off = 16; off > 0; off >>= 1) v += __shfl_xor(v, off, 32);
    return v;
}

__device__ __forceinline__ void load_xv(const float* __restrict__ x1,
                                        int b, int l, float xv[ORD][5]) {
    const float* xb = x1 + (size_t)b * ORD * LEN;
#pragma unroll
    for (int c = 0; c < ORD; ++c) {
#pragma unroll
        for (int d = 0; d < 5; ++d) {
            const int idx = l - 2 + d;
            xv[c][d] = (idx >= 0 && idx < LEN) ? xb[(size_t)c * LEN + idx] : 0.0f;
        }
    }
}

// y1 = conv k3 dil1 pad1, y2 = conv k3 dil2 pad2
__device__ __forceinline__ void conv_pair(const float xv[ORD][5],
        const float* W1, const float* B1, const float* W2, const float* B2,
        float y1[ORD], float y2[ORD]) {
#pragma unroll
    for (int o = 0; o < ORD; ++o) {
        float a1 = B1[o], a2 = B2[o];
#pragma unroll
        for (int c = 0; c < ORD; ++c) {
#pragma unroll
            for (int j = 0; j < 3; ++j) {
                a1 += W1[(o * ORD + c) * 3 + j] * xv[c][j + 1];  // l-1+j
                a2 += W2[(o * ORD + c) * 3 + j] * xv[c][2 * j];  // l-2+2j
            }
        }
        y1[o] = a1; y2[o] = a2;
    }
}

// ---- stage 1: partial (sum, sumsq) of y1 and y2 per channel ----
__global__ void k_stats1(const float* __restrict__ x1,
                         const float* __restrict__ c1w, const float* __restrict__ c1b,
                         const float* __restrict__ c2w, const float* __restrict__ c2b,
                         float* __restrict__ P1)
{
    __shared__ float W1[192], W2[192], B1[8], B2[8];
    __shared__ float red[8 * 32];
    const int tid = threadIdx.x;
    if (tid < 192) { W1[tid] = c1w[tid]; W2[tid] = c2w[tid]; }
    if (tid < 8)   { B1[tid] = c1b[tid]; B2[tid] = c2b[tid]; }
    __syncthreads();

    float st[32];
#pragma unroll
    for (int s = 0; s < 32; ++s) st[s] = 0.f;

    for (int p = blockIdx.x * blockDim.x + tid; p < NPOS; p += gridDim.x * blockDim.x) {
        const int b = p >> 13, l = p & (LEN - 1);
        float xv[ORD][5]; load_xv(x1, b, l, xv);
        float y1[ORD], y2[ORD];
        conv_pair(xv, W1, B1, W2, B2, y1, y2);
#pragma unroll
        for (int c = 0; c < ORD; ++c) {
            st[c]      += y1[c];
            st[8 + c]  += y1[c] * y1[c];
            st[16 + c] += y2[c];
            st[24 + c] += y2[c] * y2[c];
        }
    }
    const int wid = tid >> 5, lane = tid & 31;
#pragma unroll
    for (int s = 0; s < 32; ++s) {
        const float r = waveSum(st[s]);
        if (lane == 0) red[wid * 32 + s] = r;
    }
    __syncthreads();
    if (tid < 32) {
        float t = 0.f;
        for (int w = 0; w < (int)(blockDim.x >> 5); ++w) t += red[w * 32 + tid];
        P1[blockIdx.x * 32 + tid] = t;
    }
}

// ---- reduce stage 1 -> BN1/BN2 affine params ----
__global__ void k_bn12(const float* __restrict__ P1,
                       const float* g1, const float* b1,
                       const float* g2, const float* b2,
                       float* __restrict__ bn, int nblk)
{
    __shared__ float tot[32];
    const int tid = threadIdx.x;
    if (tid < 32) {
        float t = 0.f;
        for (int g = 0; g < nblk; ++g) t += P1[g * 32 + tid];
        tot[tid] = t;
    }
    __syncthreads();
    if (tid < 8) {
        const float invN = 1.0f / (float)NPOS;
        float m1 = tot[tid] * invN;
        float v1 = tot[8 + tid] * invN - m1 * m1;
        float a1 = g1[tid] * rsqrtf(v1 + 1e-5f);
        bn[tid]      = a1;
        bn[8 + tid]  = b1[tid] - a1 * m1;
        float m2 = tot[16 + tid] * invN;
        float v2 = tot[24 + tid] * invN - m2 * m2;
        float a2 = g2[tid] * rsqrtf(v2 + 1e-5f);
        bn[16 + tid] = a2;
        bn[24 + tid] = b2[tid] - a2 * m2;
    }
}

// ---- stage 2: recompute, z = pxw(relu(bn1(y1)) + relu(bn2(y2))) stats ----
__global__ void k_stats2(const float* __restrict__ x1,
                         const float* c1w, const float* c1b,
                         const float* c2w, const float* c2b,
                         const float* pxw, const float* pxb,
                         const float* __restrict__ bn,
                         float* __restrict__ P2)
{
    __shared__ float W1[192], W2[192], B1[8], B2[8], PW[64], PB[8], BN[32];
    __shared__ float red[8 * 16];
    const int tid = threadIdx.x;
    if (tid < 192) { W1[tid] = c1w[tid]; W2[tid] = c2w[tid]; }
    if (tid < 64)  PW[tid] = pxw[tid];
    if (tid < 32)  BN[tid] = bn[tid];
    if (tid < 8)   { B1[tid] = c1b[tid]; B2[tid] = c2b[tid]; PB[tid] = pxb[tid]; }
    __syncthreads();

    float st[16];
#pragma unroll
    for (int s = 0; s < 16; ++s) st[s] = 0.f;

    for (int p = blockIdx.x * blockDim.x + tid; p < NPOS; p += gridDim.x * blockDim.x) {
        const int b = p >> 13, l = p & (LEN - 1);
        float xv[ORD][5]; load_xv(x1, b, l, xv);
        float y1[ORD], y2[ORD];
        conv_pair(xv, W1, B1, W2, B2, y1, y2);
        float sarr[ORD];
#pragma unroll
        for (int c = 0; c < ORD; ++c)
            sarr[c] = fmaxf(BN[c] * y1[c] + BN[8 + c], 0.f)
                    + fmaxf(BN[16 + c] * y2[c] + BN[24 + c], 0.f);
#pragma unroll
        for (int o = 0; o < ORD; ++o) {
            float z = PB[o];
#pragma unroll
            for (int c = 0; c < ORD; ++c) z += PW[o * ORD + c] * sarr[c];
            st[o]     += z;
            st[8 + o] += z * z;
        }
    }
    const int wid = tid >> 5, lane = tid & 31;
#pragma unroll
    for (int s = 0; s < 16; ++s) {
        const float r = waveSum(st[s]);
        if (lane == 0) red[wid * 16 + s] = r;
    }
    __syncthreads();
    if (tid < 16) {
        float t = 0.f;
        for (int w = 0; w < (int)(blockDim.x >> 5); ++w) t += red[w * 16 + tid];
        P2[blockIdx.x * 16 + tid] = t;
    }
}

// ---- reduce stage 2 -> BN3 params ----
__global__ void k_bn3(const float* __restrict__ P2,
                      const float* g3, const float* b3,
                      float* __restrict__ bn, int nblk)
{
    __shared__ float tot[16];
    const int tid = threadIdx.x;
    if (tid < 16) {
        float t = 0.f;
        for (int g = 0; g < nblk; ++g) t += P2[g * 16 + tid];
        tot[tid] = t;
    }
    __syncthreads();
    if (tid < 8) {
        const float invN = 1.0f / (float)NPOS;
        float m = tot[tid] * invN;
        float v = tot[8 + tid] * invN - m * m;
        float a = g3[tid] * rsqrtf(v + 1e-5f);
        bn[32 + tid] = a;
        bn[40 + tid] = b3[tid] - a * m;
    }
}

// ---- stage 3: c = relu(bn3(z)); partial pooled sums per (b,c) ----
__global__ void k_pool(const float* __restrict__ x1,
                       const float* c1w, const float* c1b,
                       const float* c2w, const float* c2b,
                       const float* pxw, const float* pxb,
                       const float* __restrict__ bn,
                       float* __restrict__ P3)
{
    __shared__ float W1[192], W2[192], B1[8], B2[8], PW[64], PB[8], BN[48];
    __shared__ float red[32 * 8];
    const int tid = threadIdx.x;   // 1024
    if (tid < 192) { W1[tid] = c1w[tid]; W2[tid] = c2w[tid]; }
    if (tid < 64)  PW[tid] = pxw[tid];
    if (tid < 48)  BN[tid] = bn[tid];
    if (tid < 8)   { B1[tid] = c1b[tid]; B2[tid] = c2b[tid]; PB[tid] = pxb[tid]; }
    __syncthreads();

    const int b = blockIdx.x;
    const int l = blockIdx.y * blockDim.x + tid;

    float xv[ORD][5]; load_xv(x1, b, l, xv);
    float y1[ORD], y2[ORD];
    conv_pair(xv, W1, B1, W2, B2, y1, y2);
    float sarr[ORD];
#pragma unroll
    for (int c = 0; c < ORD; ++c)
        sarr[c] = fmaxf(BN[c] * y1[c] + BN[8 + c], 0.f)
                + fmaxf(BN[16 + c] * y2[c] + BN[24 + c], 0.f);
    float cv[ORD];
#pragma unroll
    for (int o = 0; o < ORD; ++o) {
        float z = PB[o];
#pragma unroll
        for (int c = 0; c < ORD; ++c) z += PW[o * ORD + c] * sarr[c];
        cv[o] = fmaxf(BN[32 + o] * z + BN[40 + o], 0.f);
    }
    const int wid = tid >> 5, lane = tid & 31;
#pragma unroll
    for (int c = 0; c < ORD; ++c) {
        const float r = waveSum(cv[c]);
        if (lane == 0) red[wid * 8 + c] = r;
    }
    __syncthreads();
    if (tid < 8) {
        float t = 0.f;
        for (int w = 0; w < 32; ++w) t += red[w * 8 + tid];
        P3[((size_t)b * 8 + blockIdx.y) * 8 + tid] = t;
    }
}

// ---- reduce stage 3 -> FC -> scalar per batch (optional clip) ----
__global__ void k_fc(const float* __restrict__ P3,
                     const float* fcw, const float* fcb,
                     float* __restrict__ outscalar, int doClip)
{
    const int b = threadIdx.x;
    if (b < BATCH) {
        float val = fcb[0];
#pragma unroll
        for (int c = 0; c < ORD; ++c) {
            float pooled = 0.f;
            for (int lb = 0; lb < 8; ++lb) pooled += P3[((size_t)b * 8 + lb) * 8 + c];
            pooled *= (1.0f / (float)LEN);
            val += pooled * fcw[c];
        }
        if (doClip) val = fminf(fmaxf(val, 1.0f), 128.0f);
        outscalar[b] = val;
    }
}

// ---- Hermite kernel coefficients: [B, ORD, KW] ----
__global__ void k_herm(const float* __restrict__ width,
                       const float* __restrict__ center,
                       float* __restrict__ herm)
{
    const int b = threadIdx.x;
    if (b >= BATCH) return;
    const float w = width[b], cen = center[b];
    float t[KW], e[KW];
#pragma unroll
    for (int j = 0; j < KW; ++j) {
        t[j] = w * ((float)j - cen);
        e[j] = expf(-0.5f * t[j] * t[j]);
    }
    float hm2[KW], hm1[KW];
    float fact = 1.0f;
    float* out = herm + (size_t)b * ORD * KW;
    for (int i = 0; i < ORD - 1; ++i) {
        float h[KW];
#pragma unroll
        for (int j = 0; j < KW; ++j) {
            if (i == 0)      h[j] = 1.0f;
            else if (i == 1) h[j] = 2.0f * t[j];
            else             h[j] = 2.0f * t[j] * hm1[j] - 2.0f * (float)(i - 1) * hm2[j];
        }
        if (i >= 1) fact *= (float)i;
        const float cst = exp2f(0.5f * (float)i) / sqrtf(PI_F * fact);
        float f[KW], nrm = 0.f;
#pragma unroll
        for (int j = 0; j < KW; ++j) { f[j] = e[j] * cst * h[j]; nrm += f[j] * f[j]; }
        nrm = fmaxf(sqrtf(nrm), 1e-12f);
#pragma unroll
        for (int j = 0; j < KW; ++j) out[i * KW + j] = f[j] / nrm;
#pragma unroll
        for (int j = 0; j < KW; ++j) { hm2[j] = hm1[j]; hm1[j] = h[j]; }
    }
#pragma unroll
    for (int j = 0; j < KW; ++j)
        out[(ORD - 1) * KW + j] = 1.0f / (1.0f + expf(-2.0f * t[j]));
}

// ---- final per-batch 9-tap, 8-channel summed conv ----
__global__ void k_out(const float* __restrict__ x1,
                      const float* __restrict__ herm,
                      const float* __restrict__ conv2_b,
                      float* __restrict__ out)
{
    __shared__ float kl[ORD * KW];
    const int b = blockIdx.x;
    const int tid = threadIdx.x;
    if (tid < ORD * KW) kl[tid] = herm[(size_t)b * ORD * KW + tid];
    __syncthreads();
    const int l = blockIdx.y * blockDim.x + tid;
    float acc = conv2_b[0];
    const float* xb = x1 + (size_t)b * ORD * LEN;
#pragma unroll
    for (int c = 0; c < ORD; ++c) {
        const float* xc = xb + (size_t)c * LEN;
#pragma unroll
        for (int j = 0; j < KW; ++j) {
            const int idx = l + j - 4;
            const float xval = (idx >= 0 && idx < LEN) ? xc[idx] : 0.0f;
            acc += xval * kl[c * KW + j];
        }
    }
    out[(size_t)b * LEN + l] = acc;
}

// =====================================================================
extern "C" void kernel_launch(void* const* d_in, const int* in_sizes, int n_in,
                              void* d_out, int out_size, void* d_ws, size_t ws_size,
                              hipStream_t stream)
{
    (void)in_sizes; (void)n_in; (void)out_size; (void)ws_size;
    const float* x       = (const float*)d_in[0];
    const float* conv1_w = (const float*)d_in[1];
    const float* conv1_b = (const float*)d_in[2];
    const float* conv2_b = (const float*)d_in[3];

    float* out = (float*)d_out;                       // [32,1,8192]
    float* x1  = out + (size_t)BATCH * LEN;           // [32,8,8192]
    float* ws  = (float*)d_ws;
    float* P1     = ws + WS_P1;
    float* P2     = ws + WS_P2;
    float* BN     = ws + WS_BN;
    float* P3     = ws + WS_P3;
    float* WIDTH  = ws + WS_WIDTH;
    float* CENTER = ws + WS_CENTER;
    float* HERM   = ws + WS_HERM;

    // 1) pointwise GEMM via fp32 WMMA: 8192 wave-tiles (32 cols each), 8 waves/block
    k_conv1<<<1024, 256, 0, stream>>>(x, conv1_w, conv1_b, x1);

    // 2) the two adaptive_r heads (dict order: c1w,c1b,c2w,c2b,g1,g2,g3,b1,b2,b3,pxw,pxb,fcw,fcb)
    for (int m = 0; m < 2; ++m) {
        const int base = 4 + m * 14;
        const float* c1w = (const float*)d_in[base + 0];
        const float* c1b = (const float*)d_in[base + 1];
        const float* c2w = (const float*)d_in[base + 2];
        const float* c2b = (const float*)d_in[base + 3];
        const float* g1  = (const float*)d_in[base + 4];
        const float* g2  = (const float*)d_in[base + 5];
        const float* g3  = (const float*)d_in[base + 6];
        const float* b1  = (const float*)d_in[base + 7];
        const float* b2  = (const float*)d_in[base + 8];
        const float* b3  = (const float*)d_in[base + 9];
        const float* pxw = (const float*)d_in[base + 10];
        const float* pxb = (const float*)d_in[base + 11];
        const float* fcw = (const float*)d_in[base + 12];
        const float* fcb = (const float*)d_in[base + 13];
        float* scal = (m == 0) ? WIDTH : CENTER;

        k_stats1<<<256, 256, 0, stream>>>(x1, c1w, c1b, c2w, c2b, P1);
        k_bn12  <<<1, 32, 0, stream>>>(P1, g1, b1, g2, b2, BN, 256);
        k_stats2<<<256, 256, 0, stream>>>(x1, c1w, c1b, c2w, c2b, pxw, pxb, BN, P2);
        k_bn3   <<<1, 32, 0, stream>>>(P2, g3, b3, BN, 256);
        k_pool  <<<dim3(32, 8), 1024, 0, stream>>>(x1, c1w, c1b, c2w, c2b, pxw, pxb, BN, P3);
        k_fc    <<<1, 32, 0, stream>>>(P3, fcw, fcb, scal, m == 1 ? 1 : 0);
    }

    // 3) Hermite coefficients and final conv
    k_herm<<<1, 32, 0, stream>>>(WIDTH, CENTER, HERM);
    k_out <<<dim3(32, 32), 256, 0, stream>>>(x1, HERM, conv2_b, out);
}